// VectorQuantization_77386720740040
// MI455X (gfx1250) — compile-verified
//
#include <hip/hip_runtime.h>
#include <hip/hip_bf16.h>

// ---------------- problem constants ----------------
#define N_TOK   65536            // 16*4096 rows
#define DDIM    256              // embedding size d
#define KEMB    1024             // number of codebook entries K
#define DECAYF  0.99f
#define EPSF    1e-5f
#define VQCOMM  0.25f

// ---------------- output layout (floats, concatenated in return order) -----
#define OFF_ZQ     0                       // 16777216
#define OFF_LOSS   16777216                // 1
#define OFF_ENEW   16777217                // 262144
#define OFF_CS     17039361                // 1024
#define OFF_EMEAN  17040385                // 262144

// ---------------- workspace layout (floats) ----------------
#define WS_ENORM   0            // [1024]
#define WS_COUNTS  1024         // [1024]
#define WS_CSN     2048         // [1024]
#define WS_LOSS    3072         // [1]
#define WS_NSUM    3073         // [1]
#define WS_EMBT    4096         // [256*1024] embedding transposed [K][d]
#define WS_ESUMT   266240       // [1024*256] embed_sum transposed [K][d]
#define WS_IDX     528384       // [65536] int

typedef __attribute__((ext_vector_type(2))) float v2f;
typedef __attribute__((ext_vector_type(8))) float v8f;

// ---------------- CDNA5 async global->LDS path (guarded) ----------------
#if defined(__gfx1250__) && __has_builtin(__builtin_amdgcn_global_load_async_to_lds_b32)
#define USE_ASYNC 1
__device__ __forceinline__ void async_g2l_b32(const float* g, float* l) {
    __builtin_amdgcn_global_load_async_to_lds_b32(
        (__attribute__((address_space(1))) int*)(unsigned long long)g,
        (__attribute__((address_space(3))) int*)(unsigned)(unsigned long long)l,
        0, 0);
}
__device__ __forceinline__ void wait_async0() {
#if __has_builtin(__builtin_amdgcn_s_wait_asynccnt)
    __builtin_amdgcn_s_wait_asynccnt(0);
#else
    asm volatile("s_wait_asynccnt 0x0" ::: "memory");
#endif
}
#else
#define USE_ASYNC 0
#endif

// ---------------------------------------------------------------------------
// zero helper
__global__ void k_zero(float* __restrict__ p, int n) {
    int i = blockIdx.x * 256 + threadIdx.x;
    if (i < n) p[i] = 0.0f;
}

// ---------------------------------------------------------------------------
// column norms ||E_k||^2  (emb is [d][K], column k strided by K)
__global__ void k_enorm(const float* __restrict__ emb, float* __restrict__ enorm) {
    int k = blockIdx.x * 256 + threadIdx.x;
    if (k < KEMB) {
        float s = 0.0f;
        #pragma unroll 8
        for (int e = 0; e < DDIM; ++e) {
            float v = emb[(size_t)e * KEMB + k];
            s += v * v;
        }
        enorm[k] = s;
    }
}

// ---------------------------------------------------------------------------
// transpose codebook: embT[k][e] = emb[e][k]  (for coalesced gathers later)
__global__ void k_transpose(const float* __restrict__ emb, float* __restrict__ embT) {
    int i = blockIdx.x * 256 + threadIdx.x;
    if (i < DDIM * KEMB) {
        int e = i / KEMB, k = i % KEMB;
        embT[(size_t)k * DDIM + e] = emb[i];
    }
}

// ---------------------------------------------------------------------------
// Fused  (-2 z.E + ||E||^2)  GEMM + running argmin, via V_WMMA_F32_16X16X4_F32.
// Block: 256 thr (8 wave32), 128 rows. Wave owns 16 rows, holds A (16x256 f32)
// in 128 VGPRs. Per iteration: TWO 16-col tiles (2 independent WMMA chains,
// 128 WMMAs), codebook chunk double-buffered in LDS with async global->LDS
// copies for the next chunk overlapping compute on the current one.
#define BPAD   258   // LDS col stride (floats): 8B-aligned, bank-conflict free
#define CCHUNK 32    // columns staged per iteration (two 16x16 WMMA tiles)
#define NITER  (KEMB / CCHUNK)

__device__ __forceinline__ void stage_chunk(const float* __restrict__ emb,
                                            float* __restrict__ buf,
                                            int colBase, int tid) {
    #pragma unroll
    for (int j = 0; j < CCHUNK; ++j) {
        int e = tid + j * 256;          // 0..8191
        int k = e >> 5;                 // 0..255
        int c = e & 31;                 // 0..31
#if USE_ASYNC
        async_g2l_b32(emb + (size_t)k * KEMB + colBase + c, buf + c * BPAD + k);
#else
        buf[c * BPAD + k] = emb[(size_t)k * KEMB + colBase + c];
#endif
    }
}

__launch_bounds__(256)
__global__ void k_argmin(const float* __restrict__ z,
                         const float* __restrict__ emb,
                         const float* __restrict__ enorm,
                         int* __restrict__ idx) {
    __shared__ __align__(16) float bsm[2][CCHUNK * BPAD];

    const int tid  = threadIdx.x;
    const int wave = tid >> 5;
    const int lane = tid & 31;
    const int half = lane >> 4;      // 0: lanes 0-15, 1: lanes 16-31
    const int l16  = lane & 15;
    const int rowBase = blockIdx.x * 128 + wave * 16;
    const int kb = half * 2;         // K sub-offset within each 16x16x4 step

    // --- A fragments for all of d=256: 64 steps x v2f (lane: M=l16, K=4t+kb+{0,1})
    v2f a[64];
    const float* zrow = z + (size_t)(rowBase + l16) * DDIM;
    #pragma unroll
    for (int t = 0; t < 64; ++t)
        a[t] = *(const v2f*)(zrow + t * 4 + kb);

    float best[8];
    int   bidx[8];
    #pragma unroll
    for (int j = 0; j < 8; ++j) { best[j] = 3.4e38f; bidx[j] = 0; }

    // prologue: stage chunk 0
    stage_chunk(emb, bsm[0], 0, tid);
#if USE_ASYNC
    wait_async0();
#endif
    __syncthreads();

    for (int it = 0; it < NITER; ++it) {
        const int cur = it & 1;

        // stage next chunk into the other buffer (overlaps with compute below)
        if (it + 1 < NITER)
            stage_chunk(emb, bsm[cur ^ 1], (it + 1) * CCHUNK, tid);

        const float* bc0 = &bsm[cur][l16 * BPAD + kb];          // cols 0..15
        const float* bc1 = &bsm[cur][(l16 + 16) * BPAD + kb];   // cols 16..31

        v8f acc0 = {0.f,0.f,0.f,0.f,0.f,0.f,0.f,0.f};
        v8f acc1 = {0.f,0.f,0.f,0.f,0.f,0.f,0.f,0.f};
        #pragma unroll
        for (int t = 0; t < 64; ++t) {
            v2f b0 = *(const v2f*)(bc0 + t * 4);
            v2f b1 = *(const v2f*)(bc1 + t * 4);
            acc0 = __builtin_amdgcn_wmma_f32_16x16x4_f32(
                       false, a[t], false, b0, (short)0, acc0, false, false);
            acc1 = __builtin_amdgcn_wmma_f32_16x16x4_f32(
                       false, a[t], false, b1, (short)0, acc1, false, false);
        }

        const int   col0 = it * CCHUNK + l16;
        const int   col1 = col0 + 16;
        const float en0  = enorm[col0];
        const float en1  = enorm[col1];
        #pragma unroll
        for (int j = 0; j < 8; ++j) {
            float dv0 = en0 - 2.0f * acc0[j];    // ||E||^2 - 2 z.E
            if (dv0 < best[j]) { best[j] = dv0; bidx[j] = col0; }
            float dv1 = en1 - 2.0f * acc1[j];
            if (dv1 < best[j]) { best[j] = dv1; bidx[j] = col1; }
        }

#if USE_ASYNC
        wait_async0();
#endif
        __syncthreads();
    }

    // cross-lane argmin within each 16-lane half (rows 0-7 / 8-15 of the tile)
    #pragma unroll
    for (int j = 0; j < 8; ++j) {
        float bv = best[j]; int bi = bidx[j];
        #pragma unroll
        for (int off = 8; off >= 1; off >>= 1) {
            float ov = __shfl_xor(bv, off, 32);
            int   oi = __shfl_xor(bi, off, 32);
            if (ov < bv || (ov == bv && oi < bi)) { bv = ov; bi = oi; }
        }
        if (l16 == 0)
            idx[rowBase + half * 8 + j] = bi;
    }
}

// ---------------------------------------------------------------------------
// gather z_q (straight-through value), segment stats, loss partial.
// One wave per row; 8 rows per block. b128 loads/stores.
__global__ void k_scatter(const float* __restrict__ z,
                          const float* __restrict__ embT,
                          const int* __restrict__ idx,
                          float* __restrict__ zq_out,
                          float* __restrict__ counts,
                          float* __restrict__ esumT,
                          float* __restrict__ lossAcc) {
    const int wave = threadIdx.x >> 5;
    const int lane = threadIdx.x & 31;
    const int n = blockIdx.x * 8 + wave;
    const int k = idx[n];

    const float* zr   = z    + (size_t)n * DDIM;
    const float* er   = embT + (size_t)k * DDIM;
    float*       orow = zq_out + (size_t)n * DDIM;
    float*       srow = esumT  + (size_t)k * DDIM;

    float lsum = 0.0f;
    #pragma unroll
    for (int i = 0; i < 2; ++i) {
        int e = lane * 4 + i * 128;
        float4 zv = *(const float4*)(zr + e);
        float4 qv = *(const float4*)(er + e);
        *(float4*)(orow + e) = qv;          // z_q_st value == z_q
        float dx = qv.x - zv.x, dy = qv.y - zv.y;
        float dz2 = qv.z - zv.z, dw = qv.w - zv.w;
        lsum += dx * dx + dy * dy + dz2 * dz2 + dw * dw;
        atomicAdd(&srow[e + 0], zv.x);      // embed_sum (transposed layout)
        atomicAdd(&srow[e + 1], zv.y);
        atomicAdd(&srow[e + 2], zv.z);
        atomicAdd(&srow[e + 3], zv.w);
    }
    #pragma unroll
    for (int off = 16; off >= 1; off >>= 1)
        lsum += __shfl_xor(lsum, off, 32);
    if (lane == 0) {
        atomicAdd(lossAcc, lsum);
        atomicAdd(&counts[k], 1.0f);
    }
}

// ---------------------------------------------------------------------------
// cluster_size EMA + total sum n (single block)
__global__ void k_cluster(const float* __restrict__ cluster_size,
                          const float* __restrict__ counts,
                          float* __restrict__ cs_out,
                          float* __restrict__ csn_ws,
                          float* __restrict__ nsum) {
    __shared__ float red[256];
    int tid = threadIdx.x;
    float s = 0.0f;
    #pragma unroll
    for (int i = 0; i < 4; ++i) {
        int k = tid + i * 256;
        float v = cluster_size[k] * DECAYF + (1.0f - DECAYF) * counts[k];
        cs_out[k] = v;
        csn_ws[k] = v;
        s += v;
    }
    red[tid] = s;
    __syncthreads();
    for (int st = 128; st > 0; st >>= 1) {
        if (tid < st) red[tid] += red[tid + st];
        __syncthreads();
    }
    if (tid == 0) nsum[0] = red[0];
}

// ---------------------------------------------------------------------------
// embedding_mean EMA + normalized embedding
__global__ void k_final(const float* __restrict__ embedding_mean,
                        const float* __restrict__ esumT,
                        const float* __restrict__ csn,
                        const float* __restrict__ nsum,
                        float* __restrict__ enew_out,
                        float* __restrict__ emean_out) {
    int i = blockIdx.x * 256 + threadIdx.x;
    if (i < DDIM * KEMB) {
        int e = i / KEMB, k = i % KEMB;
        float emn = embedding_mean[i] * DECAYF
                  + (1.0f - DECAYF) * esumT[(size_t)k * DDIM + e];
        emean_out[i] = emn;
        float nv = nsum[0];
        float cs = (csn[k] + EPSF) / (nv + (float)KEMB * EPSF) * nv;
        enew_out[i] = emn / cs;
    }
}

// ---------------------------------------------------------------------------
__global__ void k_loss_final(const float* __restrict__ lossAcc, float* __restrict__ out) {
    out[0] = VQCOMM * lossAcc[0] / (float)((size_t)N_TOK * DDIM);
}

// ---------------------------------------------------------------------------
extern "C" void kernel_launch(void* const* d_in, const int* in_sizes, int n_in,
                              void* d_out, int out_size, void* d_ws, size_t ws_size,
                              hipStream_t stream) {
    const float* z     = (const float*)d_in[0];   // [16,4096,256]
    const float* emb   = (const float*)d_in[1];   // [256,1024]
    const float* cs    = (const float*)d_in[2];   // [1024]
    const float* emean = (const float*)d_in[3];   // [256,1024]
    float* out = (float*)d_out;
    float* ws  = (float*)d_ws;

    float* enorm  = ws + WS_ENORM;
    float* counts = ws + WS_COUNTS;
    float* csn    = ws + WS_CSN;
    float* lossA  = ws + WS_LOSS;
    float* nsum   = ws + WS_NSUM;
    float* embT   = ws + WS_EMBT;
    float* esumT  = ws + WS_ESUMT;
    int*   idx    = (int*)(ws + WS_IDX);

    // zero accumulators (counts..nsum block, and esumT)
    k_zero<<<(2050 + 255) / 256, 256, 0, stream>>>(counts, 2050);
    k_zero<<<(DDIM * KEMB + 255) / 256, 256, 0, stream>>>(esumT, DDIM * KEMB);

    k_enorm<<<(KEMB + 255) / 256, 256, 0, stream>>>(emb, enorm);
    k_transpose<<<(DDIM * KEMB + 255) / 256, 256, 0, stream>>>(emb, embT);

    k_argmin<<<N_TOK / 128, 256, 0, stream>>>(z, emb, enorm, idx);

    k_scatter<<<N_TOK / 8, 256, 0, stream>>>(z, embT, idx,
                                             out + OFF_ZQ, counts, esumT, lossA);

    k_cluster<<<1, 256, 0, stream>>>(cs, counts, out + OFF_CS, csn, nsum);

    k_final<<<(DDIM * KEMB + 255) / 256, 256, 0, stream>>>(emean, esumT, csn, nsum,
                                                           out + OFF_ENEW, out + OFF_EMEAN);

    k_loss_final<<<1, 1, 0, stream>>>(lossA, out + OFF_LOSS);
}